// fusion_model_our_21646635172735
// MI455X (gfx1250) — compile-verified
//
#include <hip/hip_runtime.h>
#include <math.h>

// ---------------------------------------------------------------------------
// MI455X (gfx1250) two-branch SAGE fusion model.
// fp32 end-to-end: workload is gather/memory-bound (~1.2 GB of mostly
// L2-resident traffic vs 63 GFLOP), so V_WMMA_F32_16X16X4_F32 keeps full
// precision at no throughput cost.  Aggregation uses a CSR gather (no
// feature-wide atomics), with deterministic per-bucket ordering.
// ---------------------------------------------------------------------------

#define N_NODES 20000
#define N_EDGES 160000
#define FCNN    512
#define FSWIM   768
#define OUTF    512
#define HIDF    128

typedef __attribute__((ext_vector_type(2))) float v2f;
typedef __attribute__((ext_vector_type(8))) float v8f;

__device__ __forceinline__ float waveSum(float v) {
#pragma unroll
    for (int m = 16; m >= 1; m >>= 1) v += __shfl_xor(v, m, 32);
    return v;
}

// ------------------------------- utility -----------------------------------
__global__ void k_zero(float* __restrict__ p, long n) {
    long i = (long)blockIdx.x * blockDim.x + threadIdx.x;
    long stride = (long)gridDim.x * blockDim.x;
    for (; i < n; i += stride) p[i] = 0.0f;
}

// ----------------------------- CSR build -----------------------------------
__global__ void k_degree(const int* __restrict__ dst, int* __restrict__ deg, int e) {
    int i = blockIdx.x * blockDim.x + threadIdx.x;
    if (i < e) atomicAdd(&deg[dst[i]], 1);
}

// single-block exclusive prefix scan: row[0]=0, row[i+1]=sum(deg[0..i])
__global__ void k_scan(const int* __restrict__ deg, int* __restrict__ row, int n) {
    __shared__ int sh[256];
    __shared__ int carry;
    int t = threadIdx.x;
    if (t == 0) carry = 0;
    __syncthreads();
    for (int base = 0; base < n; base += 256) {
        int i = base + t;
        sh[t] = (i < n) ? deg[i] : 0;
        __syncthreads();
        for (int off = 1; off < 256; off <<= 1) {
            int add = (t >= off) ? sh[t - off] : 0;
            __syncthreads();
            sh[t] += add;
            __syncthreads();
        }
        if (i < n) row[i + 1] = carry + sh[t];
        __syncthreads();
        if (t == 255) carry += sh[255];
        __syncthreads();
    }
    if (t == 0) row[0] = 0;
}

// bucket fill: adjE holds EDGE INDICES (payload order fixed later by rank-sort)
__global__ void k_fill(const int* __restrict__ dst, const int* __restrict__ row,
                       int* __restrict__ cur, int* __restrict__ adjE, int e) {
    int i = blockIdx.x * blockDim.x + threadIdx.x;
    if (i < e) {
        int d = dst[i];
        int p = atomicAdd(&cur[d], 1);
        adjE[row[d] + p] = i;
    }
}

// per-node gather + mean: block of 256 per node, deterministic accumulation
// order (bucket rank-sorted by edge index in shared memory).
template <int F>
__global__ void k_gather(const float* __restrict__ x, const int* __restrict__ srcs,
                         const int* __restrict__ row, const int* __restrict__ adjE,
                         float* __restrict__ agg) {
    __shared__ int keys[256];
    __shared__ int order[256];
    int n = blockIdx.x;
    int start = row[n];
    int deg = row[n + 1] - start;
    constexpr int PER = F / 256;
    float acc[PER];
#pragma unroll
    for (int j = 0; j < PER; j++) acc[j] = 0.0f;

    if (deg <= 256) {
        if (threadIdx.x < deg) keys[threadIdx.x] = adjE[start + threadIdx.x];
        __syncthreads();
        if (threadIdx.x < deg) {
            int my = keys[threadIdx.x];
            int r = 0;
            for (int j = 0; j < deg; j++) r += (keys[j] < my);
            order[r] = srcs[my];  // source node id, deterministic slot
        }
        __syncthreads();
        for (int e = 0; e < deg; e++) {
            const float* xs = x + (long)order[e] * F;
#pragma unroll
            for (int j = 0; j < PER; j++) acc[j] += xs[threadIdx.x + 256 * j];
        }
    } else {  // astronomically unlikely with E/N = 8; unsorted fallback
        for (int e = 0; e < deg; e++) {
            const float* xs = x + (long)srcs[adjE[start + e]] * F;
#pragma unroll
            for (int j = 0; j < PER; j++) acc[j] += xs[threadIdx.x + 256 * j];
        }
    }
    float inv = 1.0f / fmaxf((float)deg, 1.0f);
#pragma unroll
    for (int j = 0; j < PER; j++)
        agg[(long)n * F + threadIdx.x + 256 * j] = acc[j] * inv;
}

// -------------------------- WMMA GEMM kernels ------------------------------
// out[M x outd] = relu?( A @ WA^T (+ X @ WX^T) (+ bias) )
// 2x2 register-blocked: one wave computes a 32x32 macro-tile (4 WMMA tiles),
// reusing each A/B fragment twice -> 1 b64 load per v_wmma_f32_16x16x4_f32.
template <int K, bool DUAL, bool BIAS, bool RELU>
__global__ void k_gemm(const float* __restrict__ A, const float* __restrict__ WA,
                       const float* __restrict__ X, const float* __restrict__ WX,
                       const float* __restrict__ bias, float* __restrict__ out,
                       int outd, int colTiles2, int totalTiles2) {
    int wid  = (blockIdx.x * blockDim.x + threadIdx.x) >> 5;
    if (wid >= totalTiles2) return;  // wave-uniform: EXEC stays all-ones
    int lane = threadIdx.x & 31;
    int tm = wid / colTiles2;
    int tn = wid - tm * colTiles2;
    int half = lane >> 4;            // K sub-pair (offsets 0/1 vs 2/3)
    int lr   = lane & 15;
    int row0 = tm * 32 + lr;
    int col0 = tn * 32 + lr;

    const float* pa0 = A  + (long)row0 * K + 2 * half;
    const float* pa1 = pa0 + (long)16 * K;
    const float* pw0 = WA + (long)col0 * K + 2 * half;
    const float* pw1 = pw0 + (long)16 * K;

    v8f a00 = {}, a01 = {}, a10 = {}, a11 = {};
#pragma unroll 2
    for (int k = 0; k < K; k += 4) {
        v2f fa0 = *(const v2f*)(pa0 + k);
        v2f fa1 = *(const v2f*)(pa1 + k);
        v2f fb0 = *(const v2f*)(pw0 + k);
        v2f fb1 = *(const v2f*)(pw1 + k);
        a00 = __builtin_amdgcn_wmma_f32_16x16x4_f32(false, fa0, false, fb0, (short)0, a00, false, false);
        a01 = __builtin_amdgcn_wmma_f32_16x16x4_f32(false, fa0, false, fb1, (short)0, a01, false, false);
        a10 = __builtin_amdgcn_wmma_f32_16x16x4_f32(false, fa1, false, fb0, (short)0, a10, false, false);
        a11 = __builtin_amdgcn_wmma_f32_16x16x4_f32(false, fa1, false, fb1, (short)0, a11, false, false);
    }
    if (DUAL) {
        const float* px0 = X  + (long)row0 * K + 2 * half;
        const float* px1 = px0 + (long)16 * K;
        const float* pv0 = WX + (long)col0 * K + 2 * half;
        const float* pv1 = pv0 + (long)16 * K;
#pragma unroll 2
        for (int k = 0; k < K; k += 4) {
            v2f fa0 = *(const v2f*)(px0 + k);
            v2f fa1 = *(const v2f*)(px1 + k);
            v2f fb0 = *(const v2f*)(pv0 + k);
            v2f fb1 = *(const v2f*)(pv1 + k);
            a00 = __builtin_amdgcn_wmma_f32_16x16x4_f32(false, fa0, false, fb0, (short)0, a00, false, false);
            a01 = __builtin_amdgcn_wmma_f32_16x16x4_f32(false, fa0, false, fb1, (short)0, a01, false, false);
            a10 = __builtin_amdgcn_wmma_f32_16x16x4_f32(false, fa1, false, fb0, (short)0, a10, false, false);
            a11 = __builtin_amdgcn_wmma_f32_16x16x4_f32(false, fa1, false, fb1, (short)0, a11, false, false);
        }
    }
    float bv0 = BIAS ? bias[col0]      : 0.0f;
    float bv1 = BIAS ? bias[col0 + 16] : 0.0f;
    // D layout: acc[i] = D[M = tileRow + i + 8*half][N = tileCol + lr]
#pragma unroll
    for (int i = 0; i < 8; i++) {
        int r0 = tm * 32 + half * 8 + i;
        int r1 = r0 + 16;
        float v00 = a00[i] + bv0, v01 = a01[i] + bv1;
        float v10 = a10[i] + bv0, v11 = a11[i] + bv1;
        if (RELU) {
            v00 = fmaxf(v00, 0.0f); v01 = fmaxf(v01, 0.0f);
            v10 = fmaxf(v10, 0.0f); v11 = fmaxf(v11, 0.0f);
        }
        out[(long)r0 * outd + col0]      = v00;
        out[(long)r0 * outd + col0 + 16] = v01;
        out[(long)r1 * outd + col0]      = v10;
        out[(long)r1 * outd + col0 + 16] = v11;
    }
}

// ------------------------------ LayerNorm ----------------------------------
__global__ void k_layernorm(float* __restrict__ h, const float* __restrict__ g,
                            const float* __restrict__ b) {
    int wid  = (blockIdx.x * blockDim.x + threadIdx.x) >> 5;
    int lane = threadIdx.x & 31;
    if (wid >= N_NODES) return;
    float* row = h + (long)wid * OUTF;
    float vals[16];
    float s = 0.0f;
#pragma unroll
    for (int i = 0; i < 16; i++) { vals[i] = row[lane + 32 * i]; s += vals[i]; }
    float mean = waveSum(s) * (1.0f / OUTF);
    float v = 0.0f;
#pragma unroll
    for (int i = 0; i < 16; i++) { float d = vals[i] - mean; v += d * d; }
    v = waveSum(v) * (1.0f / OUTF);
    float inv = rsqrtf(v + 1e-5f);
#pragma unroll
    for (int i = 0; i < 16; i++) {
        int f = lane + 32 * i;
        row[f] = g[f] * (vals[i] - mean) * inv + b[f];
    }
}

// ------------------------------ gate MLP -----------------------------------
__global__ void k_gate(const float* __restrict__ z, const float* __restrict__ b1,
                       const float* __restrict__ w2, const float* __restrict__ b2,
                       const float* __restrict__ wp, float* __restrict__ gate) {
    int wid  = (blockIdx.x * blockDim.x + threadIdx.x) >> 5;
    int lane = threadIdx.x & 31;
    if (wid >= N_NODES) return;
    const float* zr = z + (long)wid * HIDF;
    float dot = 0.0f;
#pragma unroll
    for (int i = 0; i < 4; i++) {
        int hx = lane + 32 * i;
        float v = zr[hx] + b1[hx];
        if (wp) v += wp[hx];
        v = fmaxf(v, 0.0f);
        dot += v * w2[hx];
    }
    dot = waveSum(dot);
    if (lane == 0) gate[wid] = dot + b2[0];
}

// --------------------------- softmax reductions ----------------------------
__global__ void k_max_partial(const float* __restrict__ gate, float* __restrict__ red) {
    __shared__ float sh[256];
    float m = -1e30f;
    for (int i = blockIdx.x * blockDim.x + threadIdx.x; i < N_NODES;
         i += gridDim.x * blockDim.x)
        m = fmaxf(m, gate[i]);
    sh[threadIdx.x] = m; __syncthreads();
    for (int s = 128; s > 0; s >>= 1) {
        if (threadIdx.x < s) sh[threadIdx.x] = fmaxf(sh[threadIdx.x], sh[threadIdx.x + s]);
        __syncthreads();
    }
    if (threadIdx.x == 0) red[blockIdx.x] = sh[0];
}

__global__ void k_max_final(const float* __restrict__ red, float* __restrict__ stat) {
    __shared__ float sh[256];
    sh[threadIdx.x] = red[threadIdx.x]; __syncthreads();
    for (int s = 128; s > 0; s >>= 1) {
        if (threadIdx.x < s) sh[threadIdx.x] = fmaxf(sh[threadIdx.x], sh[threadIdx.x + s]);
        __syncthreads();
    }
    if (threadIdx.x == 0) stat[0] = sh[0];
}

__global__ void k_expsum_partial(const float* __restrict__ gate, const float* __restrict__ stat,
                                 float* __restrict__ red) {
    __shared__ float sh[256];
    float gmax = stat[0];
    float s = 0.0f;
    for (int i = blockIdx.x * blockDim.x + threadIdx.x; i < N_NODES;
         i += gridDim.x * blockDim.x)
        s += __expf(gate[i] - gmax);
    sh[threadIdx.x] = s; __syncthreads();
    for (int st = 128; st > 0; st >>= 1) {
        if (threadIdx.x < st) sh[threadIdx.x] += sh[threadIdx.x + st];
        __syncthreads();
    }
    if (threadIdx.x == 0) red[blockIdx.x] = sh[0];
}

__global__ void k_expsum_final(const float* __restrict__ red, float* __restrict__ stat) {
    __shared__ float sh[256];
    sh[threadIdx.x] = red[threadIdx.x]; __syncthreads();
    for (int s = 128; s > 0; s >>= 1) {
        if (threadIdx.x < s) sh[threadIdx.x] += sh[threadIdx.x + s];
        __syncthreads();
    }
    if (threadIdx.x == 0) stat[1] = sh[0];
}

// deterministic two-stage weighted-sum pooling
#define POOL_CHUNK 64
#define POOL_BLOCKS ((N_NODES + POOL_CHUNK - 1) / POOL_CHUNK)  // 313
__global__ void k_pool_part(const float* __restrict__ gate, const float* __restrict__ stat,
                            const float* __restrict__ h, float* __restrict__ part) {
    float gmax = stat[0], invZ = 1.0f / stat[1];
    int n0 = blockIdx.x * POOL_CHUNK;
    int n1 = min(n0 + POOL_CHUNK, N_NODES);
    int f0 = threadIdx.x, f1 = threadIdx.x + 256;
    float a0 = 0.0f, a1 = 0.0f;
    for (int n = n0; n < n1; n++) {
        float w = __expf(gate[n] - gmax) * invZ;
        const float* hr = h + (long)n * OUTF;
        a0 += w * hr[f0];
        a1 += w * hr[f1];
    }
    part[(long)blockIdx.x * OUTF + f0] = a0;
    part[(long)blockIdx.x * OUTF + f1] = a1;
}

__global__ void k_pool_final(const float* __restrict__ part, float* __restrict__ pvec) {
    int f = blockIdx.x * blockDim.x + threadIdx.x;
    if (f < OUTF) {
        float s = 0.0f;
        for (int b = 0; b < POOL_BLOCKS; b++) s += part[(long)b * OUTF + f];
        pvec[f] = s;
    }
}

// wp[j] = p . w1[j,:]
__global__ void k_wp(const float* __restrict__ p, const float* __restrict__ w1,
                     float* __restrict__ wp) {
    int j = threadIdx.x;
    if (j < HIDF) {
        float s = 0.0f;
        const float* wr = w1 + (long)j * OUTF;
        for (int f = 0; f < OUTF; f++) s += p[f] * wr[f];
        wp[j] = s;
    }
}

// ------------------------------- head --------------------------------------
__global__ void k_head(const float* __restrict__ pc1, const float* __restrict__ pc2p,
                       const float* __restrict__ ps1, const float* __restrict__ ps2p,
                       const float* __restrict__ l1c_w, const float* __restrict__ l1c_b,
                       const float* __restrict__ l1s_w, const float* __restrict__ l1s_b,
                       const float* __restrict__ nc_g, const float* __restrict__ nc_b,
                       const float* __restrict__ ns_g, const float* __restrict__ ns_b,
                       const float* __restrict__ fc_w, const float* __restrict__ fc_b,
                       float* __restrict__ out) {
    __shared__ float vc[OUTF], vs[OUTF], hc[HIDF], hs[HIDF], red[256];
    int t = threadIdx.x;
    for (int f = t; f < OUTF; f += 256) {
        vc[f] = pc1[f] + pc2p[f];  // p2 = pooled2 + p1 (softmax sums to 1)
        vs[f] = ps1[f] + ps2p[f];
    }
    __syncthreads();
    float sc = 0.0f, ss = 0.0f;
    for (int f = t; f < OUTF; f += 256) { sc += vc[f] * vc[f]; ss += vs[f] * vs[f]; }
    red[t] = sc; __syncthreads();
    for (int s = 128; s > 0; s >>= 1) { if (t < s) red[t] += red[t + s]; __syncthreads(); }
    float invc = 1.0f / fmaxf(sqrtf(red[0]), 1e-12f); __syncthreads();
    red[t] = ss; __syncthreads();
    for (int s = 128; s > 0; s >>= 1) { if (t < s) red[t] += red[t + s]; __syncthreads(); }
    float invs = 1.0f / fmaxf(sqrtf(red[0]), 1e-12f); __syncthreads();
    if (t < HIDF) {
        float a = 0.0f, b = 0.0f;
        const float* wc = l1c_w + (long)t * OUTF;
        const float* wsr = l1s_w + (long)t * OUTF;
        for (int f = 0; f < OUTF; f++) { a += vc[f] * wc[f]; b += vs[f] * wsr[f]; }
        hc[t] = fmaxf(a * invc + l1c_b[t], 0.0f);
        hs[t] = fmaxf(b * invs + l1s_b[t], 0.0f);
    }
    __syncthreads();
    red[t] = (t < HIDF) ? hc[t] : 0.0f; __syncthreads();
    for (int s = 128; s > 0; s >>= 1) { if (t < s) red[t] += red[t + s]; __syncthreads(); }
    float mc = red[0] * (1.0f / HIDF); __syncthreads();
    red[t] = (t < HIDF) ? (hc[t] - mc) * (hc[t] - mc) : 0.0f; __syncthreads();
    for (int s = 128; s > 0; s >>= 1) { if (t < s) red[t] += red[t + s]; __syncthreads(); }
    float ic = rsqrtf(red[0] * (1.0f / HIDF) + 1e-5f); __syncthreads();
    red[t] = (t < HIDF) ? hs[t] : 0.0f; __syncthreads();
    for (int s = 128; s > 0; s >>= 1) { if (t < s) red[t] += red[t + s]; __syncthreads(); }
    float ms = red[0] * (1.0f / HIDF); __syncthreads();
    red[t] = (t < HIDF) ? (hs[t] - ms) * (hs[t] - ms) : 0.0f; __syncthreads();
    for (int s = 128; s > 0; s >>= 1) { if (t < s) red[t] += red[t + s]; __syncthreads(); }
    float is = rsqrtf(red[0] * (1.0f / HIDF) + 1e-5f); __syncthreads();
    if (t < HIDF) {
        hc[t] = nc_g[t] * (hc[t] - mc) * ic + nc_b[t];
        hs[t] = ns_g[t] * (hs[t] - ms) * is + ns_b[t];
    }
    __syncthreads();
    if (t < 3) {
        float a = fc_b[t];
        const float* w = fc_w + (long)t * (2 * HIDF);
        for (int j = 0; j < HIDF; j++) a += hc[j] * w[j] + hs[j] * w[HIDF + j];
        out[t] = 1.0f / (1.0f + __expf(-a));
    }
}

// ---------------------------------------------------------------------------
extern "C" void kernel_launch(void* const* d_in, const int* in_sizes, int n_in,
                              void* d_out, int out_size, void* d_ws, size_t ws_size,
                              hipStream_t stream) {
    const float* x_cnn  = (const float*)d_in[0];
    const float* x_swim = (const float*)d_in[1];
    const int*   e_c    = (const int*)d_in[2];
    const int*   e_s    = (const int*)d_in[3];
    const float* wl_c   = (const float*)d_in[4];
    const float* bl_c   = (const float*)d_in[5];
    const float* wr_c   = (const float*)d_in[6];
    const float* wl_s   = (const float*)d_in[7];
    const float* bl_s   = (const float*)d_in[8];
    const float* wr_s   = (const float*)d_in[9];
    const float* ln_g   = (const float*)d_in[10];
    const float* ln_b   = (const float*)d_in[11];
    const float* ac1_w  = (const float*)d_in[12];
    const float* ac1_b  = (const float*)d_in[13];
    const float* ac2_w  = (const float*)d_in[14];
    const float* ac2_b  = (const float*)d_in[15];
    const float* as1_w  = (const float*)d_in[16];
    const float* as1_b  = (const float*)d_in[17];
    const float* as2_w  = (const float*)d_in[18];
    const float* as2_b  = (const float*)d_in[19];
    const float* l1c_w  = (const float*)d_in[20];
    const float* l1c_b  = (const float*)d_in[21];
    const float* l1s_w  = (const float*)d_in[22];
    const float* l1s_b  = (const float*)d_in[23];
    const float* nc_g   = (const float*)d_in[24];
    const float* nc_b   = (const float*)d_in[25];
    const float* ns_g   = (const float*)d_in[26];
    const float* ns_b   = (const float*)d_in[27];
    const float* fc_w   = (const float*)d_in[28];
    const float* fc_b   = (const float*)d_in[29];

    // ---- workspace carve ----
    float* ws     = (float*)d_ws;
    float* agg_c  = ws;                                   // N*512
    float* agg_s  = agg_c + (long)N_NODES * FCNN;         // N*768
    float* h_cnn  = agg_s + (long)N_NODES * FSWIM;        // N*512
    float* h_swim = h_cnn + (long)N_NODES * OUTF;         // N*512
    float* gate   = h_swim + (long)N_NODES * OUTF;        // N
    float* red    = gate + N_NODES;                       // 256
    float* stat   = red + 256;                            // [gmax, Z]
    float* p_c1   = stat + 8;                             // 512 x4
    float* p_s1   = p_c1 + OUTF;
    float* p_c2   = p_s1 + OUTF;
    float* p_s2   = p_c2 + OUTF;
    float* wp_c   = p_s2 + OUTF;                          // 128 x2
    float* wp_s   = wp_c + HIDF;
    float* part   = wp_s + HIDF;                          // POOL_BLOCKS*512
    int*   ibase  = (int*)(part + (long)POOL_BLOCKS * OUTF);
    int*   deg_c  = ibase;                                // N
    int*   deg_s  = deg_c + N_NODES;                      // N
    int*   cur_c  = deg_s + N_NODES;                      // N
    int*   cur_s  = cur_c + N_NODES;                      // N
    int*   row_c  = cur_s + N_NODES;                      // N+1
    int*   row_s  = row_c + (N_NODES + 1);                // N+1
    int*   adj_c  = row_s + (N_NODES + 1);                // E (edge indices)
    int*   adj_s  = adj_c + N_EDGES;                      // E
    // z buffers alias agg_c (dead after the SAGE GEMMs)
    float* z_c    = agg_c;                                // N*128
    float* z_s    = agg_c + (long)N_NODES * HIDF;         // N*128

    // 1) zero int accumulators (deg_c|deg_s|cur_c|cur_s contiguous; 0.0f == 0x0)
    k_zero<<<128, 256, 0, stream>>>((float*)deg_c, 4L * N_NODES);

    // 2) CSR build for both edge lists (src = row0, dst = row1)
    k_degree<<<(N_EDGES + 255) / 256, 256, 0, stream>>>(e_c + N_EDGES, deg_c, N_EDGES);
    k_degree<<<(N_EDGES + 255) / 256, 256, 0, stream>>>(e_s + N_EDGES, deg_s, N_EDGES);
    k_scan<<<1, 256, 0, stream>>>(deg_c, row_c, N_NODES);
    k_scan<<<1, 256, 0, stream>>>(deg_s, row_s, N_NODES);
    k_fill<<<(N_EDGES + 255) / 256, 256, 0, stream>>>(e_c + N_EDGES, row_c, cur_c, adj_c, N_EDGES);
    k_fill<<<(N_EDGES + 255) / 256, 256, 0, stream>>>(e_s + N_EDGES, row_s, cur_s, adj_s, N_EDGES);

    // 3) deterministic mean-gather (no feature-wide atomics; L2-resident reads)
    k_gather<FCNN><<<N_NODES, 256, 0, stream>>>(x_cnn, e_c, row_c, adj_c, agg_c);
    k_gather<FSWIM><<<N_NODES, 256, 0, stream>>>(x_swim, e_s, row_s, adj_s, agg_s);

    // 4) SAGE dual-GEMMs (fp32 WMMA, 2x2 macro-tiles): h = relu(agg@Wl^T+bl+x@Wr^T)
    // macro-tiles = (20000/32)*(512/32) = 625*16 = 10000 -> 1250 blocks
    k_gemm<FCNN, true, true, true><<<1250, 256, 0, stream>>>(
        agg_c, wl_c, x_cnn, wr_c, bl_c, h_cnn, OUTF, 16, 10000);
    k_gemm<FSWIM, true, true, true><<<1250, 256, 0, stream>>>(
        agg_s, wl_s, x_swim, wr_s, bl_s, h_swim, OUTF, 16, 10000);

    // 5) swim branch LayerNorm (after relu)
    k_layernorm<<<N_NODES / 8, 256, 0, stream>>>(h_swim, ln_g, ln_b);

    // 6) gate hidden GEMMs z = h @ W1^T ; macro-tiles = 625*4 = 2500 -> 313 blocks
    k_gemm<OUTF, false, false, false><<<313, 256, 0, stream>>>(
        h_cnn, ac1_w, nullptr, nullptr, nullptr, z_c, HIDF, 4, 2500);
    k_gemm<OUTF, false, false, false><<<313, 256, 0, stream>>>(
        h_swim, as1_w, nullptr, nullptr, nullptr, z_s, HIDF, 4, 2500);

    auto run_pool = [&](const float* z, const float* b1, const float* w2, const float* b2,
                        const float* wp, const float* h, float* pvec) {
        k_gate<<<N_NODES / 8, 256, 0, stream>>>(z, b1, w2, b2, wp, gate);
        k_max_partial<<<256, 256, 0, stream>>>(gate, red);
        k_max_final<<<1, 256, 0, stream>>>(red, stat);
        k_expsum_partial<<<256, 256, 0, stream>>>(gate, stat, red);
        k_expsum_final<<<1, 256, 0, stream>>>(red, stat);
        k_pool_part<<<POOL_BLOCKS, 256, 0, stream>>>(gate, stat, h, part);
        k_pool_final<<<2, 256, 0, stream>>>(part, pvec);
    };

    // 7) first pooling pass
    run_pool(z_c, ac1_b, ac2_w, ac2_b, nullptr, h_cnn, p_c1);
    run_pool(z_s, as1_b, as2_w, as2_b, nullptr, h_swim, p_s1);

    // 8) second pass: gate2 = relu(z + p1@W1^T + b1).w2 + b2 ; p2 = pooled + p1
    k_wp<<<1, 128, 0, stream>>>(p_c1, ac1_w, wp_c);
    k_wp<<<1, 128, 0, stream>>>(p_s1, as1_w, wp_s);
    run_pool(z_c, ac1_b, ac2_w, ac2_b, wp_c, h_cnn, p_c2);
    run_pool(z_s, as1_b, as2_w, as2_b, wp_s, h_swim, p_s2);

    // 9) head
    k_head<<<1, 256, 0, stream>>>(p_c1, p_c2, p_s1, p_s2,
                                  l1c_w, l1c_b, l1s_w, l1s_b,
                                  nc_g, nc_b, ns_g, ns_b,
                                  fc_w, fc_b, (float*)d_out);
}